// MtMEvaluation_67353677135975
// MI455X (gfx1250) — compile-verified
//
#include <hip/hip_runtime.h>
#include <math.h>

// ---------------------------------------------------------------------------
// PnP refine (5 Gauss-Newton iters) for 131072 independent tiny problems.
// Wave-per-problem WMMA (V_WMMA_F32_16X16X4_F32) for Q = P^T * MtM * P,
// thread-per-problem f64 VALU for the 6x6 solve + se(3) exp.
// All LDS exchange is wave-local -> no block barriers, only s_wait_dscnt.
// ---------------------------------------------------------------------------

typedef float v2f __attribute__((ext_vector_type(2)));
typedef float v8f __attribute__((ext_vector_type(8)));

#define NPB   128   // problems (and threads) per block
#define NWAVE 4     // waves per block

// Wave-local LDS store->load ordering: CDNA5 LDS ops from one wave complete
// in order (DScnt); we only need to stop the compiler from reordering and
// drain DScnt before dependent cross-lane reads.
__device__ __forceinline__ void wave_lds_fence() {
  __builtin_amdgcn_wave_barrier();
  asm volatile("s_wait_dscnt 0x0" ::: "memory");
  __builtin_amdgcn_wave_barrier();
}

// P(T): [13x7] matrix, zero outside r<13, c<7 (padded to 16x16).
__device__ __forceinline__ float pofT(int r, int c, const float* T) {
  if (r < 9) {
    int i = r / 3, k = r - 3 * i;
    float a = T[i*4 + 0], b = T[i*4 + 1], cc = T[i*4 + 2];
    if (k == 0) { if (c == 1) return -cc; if (c == 2) return b;  if (c == 6) return a;  return 0.f; }
    if (k == 1) { if (c == 0) return cc;  if (c == 2) return -a; if (c == 6) return b;  return 0.f; }
    /* k==2 */  { if (c == 0) return -b;  if (c == 1) return a;  if (c == 6) return cc; return 0.f; }
  }
  if (r == 9) return (c == 6) ? 1.f : 0.f;
  if (r < 13) { int i = r - 10; if (c >= 3 && c <= 6) return T[i*4 + (c - 3)]; return 0.f; }
  return 0.f;
}

__global__ __launch_bounds__(NPB) void pnp_refine_kernel(
    const float* __restrict__ MTMs,  // [NP,13,13] f32 (88.6 MB -> L2 resident)
    const float* __restrict__ Ts,    // [NP,4,4]   f32
    double* __restrict__ outT,       // [NP,16]    f64
    double* __restrict__ outOm,      // [NP,36]    f64
    int NP)
{
  __shared__ float sTf[NPB * 16];     // published f32 poses (wave-local use)
  __shared__ float sP[NWAVE][256];    // padded 16x16 P tile per wave
  __shared__ float sX[NWAVE][256];    // padded 16x16 X = MtM*P tile per wave
  __shared__ float sQ[NPB * 56];      // Q[0:7,0:7] per problem (stride 8)

  const int tid  = threadIdx.x;
  const int lane = tid & 31;
  const int w    = tid >> 5;
  const int gp   = blockIdx.x * NPB + tid;   // problem owned for scalar phase

  // Pose state kept per-thread in f64 across the 5 iterations.
  double Tr[16];
  if (gp < NP) {
    #pragma unroll
    for (int k = 0; k < 16; ++k) Tr[k] = (double)Ts[(size_t)gp * 16 + k];
  } else {
    #pragma unroll
    for (int k = 0; k < 16; ++k) Tr[k] = (k % 5 == 0) ? 1.0 : 0.0;
  }

  const int m    = lane & 15;            // M-row (A) / N-col (B) index
  const int koff = (lane >= 16) ? 2 : 0; // K split across lane halves

  for (int it = 0; it < 5; ++it) {
    // ---- publish pose (f32); consumed only by this thread's own wave ----
    #pragma unroll
    for (int k = 0; k < 16; ++k) sTf[tid * 16 + k] = (float)Tr[k];
    wave_lds_fence();

    // ---- Phase A: wave computes Q for its 32 problems ----
    for (int s = 0; s < 32; ++s) {
      const int pl  = w * 32 + s;
      const int gpw = blockIdx.x * NPB + pl;
      const int gpc = (gpw < NP) ? gpw : (NP - 1);
      const float* Tt = &sTf[pl * 16];

      wave_lds_fence();  // prev step's sP/sX reads done before overwrite

      // Build padded P (rows>=13, cols>=7 are zero), 8 entries per lane.
      #pragma unroll
      for (int i = 0; i < 8; ++i) {
        int e = i * 32 + lane;
        sP[w][e] = pofT(e >> 4, e & 15, Tt);
      }
      wave_lds_fence();

      // A-operand slices of MtM: unconditional, address-clamped loads.
      // Garbage in rows/cols >=13 is annihilated by the zero padding of P,
      // so no predication is needed -- only in-bounds addresses.
      const float* Mp = MTMs + (size_t)gpc * 169;
      v2f a[4];
      #pragma unroll
      for (int j = 0; j < 4; ++j) {
        int i0 = m * 13 + (4 * j + koff);
        int i1 = i0 + 1;
        i0 = (i0 > 168) ? 168 : i0;
        i1 = (i1 > 168) ? 168 : i1;
        a[j].x = Mp[i0];
        a[j].y = Mp[i1];
      }

      // Step 1: X = MtM * P
      v8f x = {0.f, 0.f, 0.f, 0.f, 0.f, 0.f, 0.f, 0.f};
      #pragma unroll
      for (int j = 0; j < 4; ++j) {
        const int k0 = 4 * j + koff;
        v2f b;
        b.x = sP[w][(k0    ) * 16 + m];
        b.y = sP[w][(k0 + 1) * 16 + m];
        x = __builtin_amdgcn_wmma_f32_16x16x4_f32(false, a[j], false, b,
                                                  (short)0, x, false, false);
      }
      // Deposit X (C/D layout: vgpr i -> rows i / i+8) to LDS.
      #pragma unroll
      for (int i = 0; i < 8; ++i) {
        int row = (lane < 16) ? i : i + 8;
        sX[w][row * 16 + m] = x[i];
      }
      wave_lds_fence();

      // Step 2: Q = P^T * X  (A = P^T read transposed from LDS, B = X)
      v8f q = {0.f, 0.f, 0.f, 0.f, 0.f, 0.f, 0.f, 0.f};
      #pragma unroll
      for (int j = 0; j < 4; ++j) {
        const int k0 = 4 * j + koff;
        v2f a2, b2;
        a2.x = sP[w][(k0    ) * 16 + m];   // P^T[m][k] = P[k][m]
        a2.y = sP[w][(k0 + 1) * 16 + m];
        b2.x = sX[w][(k0    ) * 16 + m];
        b2.y = sX[w][(k0 + 1) * 16 + m];
        q = __builtin_amdgcn_wmma_f32_16x16x4_f32(false, a2, false, b2,
                                                  (short)0, q, false, false);
      }
      // Keep Q[0:7, 0:7] (vgpr i -> row i for lanes 0..15).
      if (lane < 8) {
        #pragma unroll
        for (int i = 0; i < 7; ++i) sQ[pl * 56 + i * 8 + lane] = q[i];
      }
    }
    wave_lds_fence();

    // ---- Phase B: per-thread f64 regularized solve + se(3) exp ----
    if (gp < NP) {
      const float* Qp = &sQ[tid * 56];

      const double q33 = (double)Qp[3*8+3], q44 = (double)Qp[4*8+4], q55 = (double)Qp[5*8+5];
      const double trsum = q33 + q44 + q55;
      const double regT = 1e-8 * ((trsum > 1.0) ? trsum : 1.0);

      double Aug[6][7];
      double mx = -1e300;
      #pragma unroll
      for (int i = 0; i < 6; ++i) {
        #pragma unroll
        for (int j = 0; j < 6; ++j) {
          double v = (double)Qp[i*8 + j] + ((i == j) ? (i < 3 ? 10.0 : regT) : 0.0);
          Aug[i][j] = v;
          mx = (v > mx) ? v : mx;
        }
        Aug[i][6] = -(double)Qp[i*8 + 6];
      }
      const double reg = 1e-5 * (1.0 + mx);   // reference's _inv regularizer
      #pragma unroll
      for (int i = 0; i < 6; ++i) Aug[i][i] += reg;

      // Gaussian elimination (heavily diagonally regularized PSD -> no pivot).
      #pragma unroll
      for (int k = 0; k < 6; ++k) {
        double ip = 1.0 / Aug[k][k];
        #pragma unroll
        for (int j = 0; j < 7; ++j) Aug[k][j] *= ip;
        #pragma unroll
        for (int i2 = 0; i2 < 6; ++i2) {
          if (i2 > k) {
            double f = Aug[i2][k];
            #pragma unroll
            for (int j = 0; j < 7; ++j) Aug[i2][j] -= f * Aug[k][j];
          }
        }
      }
      double v[6];
      #pragma unroll
      for (int i = 5; i >= 0; --i) {
        double sv = Aug[i][6];
        #pragma unroll
        for (int j = 0; j < 6; ++j) { if (j > i) sv -= Aug[i][j] * v[j]; }
        v[i] = sv;
      }

      // Closed-form expm of the se(3) hat matrix (exact, matches _expm).
      const double wx = v[0], wy = v[1], wz = v[2];
      const double tx = v[3], ty = v[4], tz = v[5];
      const double th2 = wx*wx + wy*wy + wz*wz;
      double Ac, Bc, Cc;
      if (th2 > 1e-14) {
        double th = sqrt(th2);
        Ac = sin(th) / th;
        Bc = (1.0 - cos(th)) / th2;
        Cc = (1.0 - Ac) / th2;
      } else {
        Ac = 1.0 - th2 * (1.0 / 6.0);
        Bc = 0.5 - th2 * (1.0 / 24.0);
        Cc = (1.0 / 6.0) - th2 * (1.0 / 120.0);
      }
      const double W[3][3] = {{0.0, -wz, wy}, {wz, 0.0, -wx}, {-wy, wx, 0.0}};
      const double wv[3] = {wx, wy, wz};
      const double tt[3] = {tx, ty, tz};
      double E[4][4];
      #pragma unroll
      for (int i = 0; i < 3; ++i) {
        double vt = 0.0;
        #pragma unroll
        for (int j = 0; j < 3; ++j) {
          double ww = wv[i] * wv[j];
          E[i][j] = ((i == j) ? (1.0 - Bc * th2) : 0.0) + Ac * W[i][j] + Bc * ww;
          double Vij = ((i == j) ? (1.0 - Cc * th2) : 0.0) + Bc * W[i][j] + Cc * ww;
          vt += Vij * tt[j];
        }
        E[i][3] = vt;
      }
      E[3][0] = 0.0; E[3][1] = 0.0; E[3][2] = 0.0; E[3][3] = 1.0;

      double Tn[16];
      #pragma unroll
      for (int i = 0; i < 4; ++i)
        #pragma unroll
        for (int j = 0; j < 4; ++j)
          Tn[i*4 + j] = Tr[i*4+0]*E[0][j] + Tr[i*4+1]*E[1][j]
                      + Tr[i*4+2]*E[2][j] + Tr[i*4+3]*E[3][j];

      if (it == 4) {
        #pragma unroll
        for (int k = 0; k < 16; ++k) outT[(size_t)gp * 16 + k] = Tn[k];
        const double tzv = Tn[2*4 + 3];
        const double sc = 1.0 / (tzv * tzv + 1e-8);
        double* Op = outOm + (size_t)gp * 36;
        #pragma unroll
        for (int i = 0; i < 6; ++i)
          #pragma unroll
          for (int j = 0; j < 6; ++j) {
            double val = (double)Qp[i*8 + j] + ((i == j) ? (i < 3 ? 10.0 : regT) : 0.0);
            Op[i*6 + j] = val * sc;
          }
      }
      #pragma unroll
      for (int k = 0; k < 16; ++k) Tr[k] = Tn[k];
    }
    wave_lds_fence();
  }
}

extern "C" void kernel_launch(void* const* d_in, const int* in_sizes, int n_in,
                              void* d_out, int out_size, void* d_ws, size_t ws_size,
                              hipStream_t stream) {
  const float* MTMs = (const float*)d_in[0];   // [B,C,13,13] f32
  const float* Ts   = (const float*)d_in[1];   // [B,C,4,4]   f32
  const int NP = in_sizes[1] / 16;             // B*C = 131072

  double* dout  = (double*)d_out;              // reference returns (T, Omega) f64
  double* outT  = dout;
  double* outOm = dout + (size_t)NP * 16;

  const int blocks = (NP + NPB - 1) / NPB;
  pnp_refine_kernel<<<blocks, NPB, 0, stream>>>(MTMs, Ts, outT, outOm, NP);
}